// LSTM_44513041056127
// MI455X (gfx1250) — compile-verified
//
#include <hip/hip_runtime.h>
#include <hip/hip_bf16.h>

// ---------------------------------------------------------------------------
// Bidirectional 2-layer LSTM on CDNA5 (gfx1250), native fp32 WMMA
// (V_WMMA_F32_16X16X4_F32), fully fused per-step kernel:
//   gates GEMM (packed-fragment coalesced loads)  ->  in-register cell update
// One kernel launch per timestep; packed h double-buffered across steps.
//
// Sizes: I=256, H=512, L=2, D=2, O=8, S=512, B=64, G=4H=2048.
// Per step & direction:  gates[64,2048] = [x_t , h_{t-1}] @ [W_ih ; W_hh]^T + b
//
// Packed "chunk" layouts (K split into chunks of 4, half = K-pair within chunk):
//   B (weights): wpk[kc][half][n][e]  = W[n][4*kc + 2*half + e]   (n = 0..2047)
//       -> lane's v2f at wpk + half*4096 + 2*n + k*2048  (lanes consecutive n)
//   A (x / h):   pa[kc][half][m][e]   = X[m][4*kc + 2*half + e]   (m = 0..63)
//       -> lane's v2f at pa + half*128 + 2*m + k*64      (lanes consecutive m)
//
// Tile mapping: wave (mtile 0..3, jt 0..31) owns batch rows mtile*16..+15 and
// hidden cols jt*16..+15 FOR ALL FOUR GATES (n = g*512 + jt*16 + lm), so the
// LSTM nonlinearity is applied entirely in registers after the K loop.
// ---------------------------------------------------------------------------

#define LSTM_S 512
#define LSTM_B 64
#define LSTM_I 256
#define LSTM_H 512
#define LSTM_G 2048   // 4*H

typedef float v2f __attribute__((ext_vector_type(2)));
typedef float v8f __attribute__((ext_vector_type(8)));

__device__ __forceinline__ v2f ld2(const float* p) {
    return *reinterpret_cast<const v2f*>(p);
}

// ---------------------------------------------------------------------------
// One-time weight pack: wcat = [W_ih ; W_hh] along K, into chunk layout.
// out[(k>>2)*8192 + ((k>>1)&1)*4096 + 2n + (k&1)] = wcat[n][k]
// ---------------------------------------------------------------------------
__global__ __launch_bounds__(256) void lstm_pack_w(
    const float* __restrict__ wih, const float* __restrict__ whh,
    int K1, int Ktot, float* __restrict__ out)
{
    const size_t idx = (size_t)blockIdx.x * 256 + threadIdx.x; // n*Ktot + k
    const int k = (int)(idx % Ktot);
    const int n = (int)(idx / Ktot);
    const float v = (k < K1) ? wih[(size_t)n * K1 + k]
                             : whh[(size_t)n * (Ktot - K1) + (k - K1)];
    out[(size_t)(k >> 2) * 8192 + (size_t)((k >> 1) & 1) * 4096 + 2 * n + (k & 1)] = v;
}

// ---------------------------------------------------------------------------
// Per-layer h0/c0 init: h -> packed A-fragment layout, c -> [dir][j][b].
// ---------------------------------------------------------------------------
__global__ __launch_bounds__(256) void lstm_init_hc(
    const float* __restrict__ h0, const float* __restrict__ c0, // [2][B][H] planes
    float* __restrict__ ph, float* __restrict__ ct)
{
    const int idx = blockIdx.x * 256 + threadIdx.x;  // 0..65535
    const int dir = idx >> 15;
    const int r = idx & 32767;
    const int b = r >> 9;
    const int j = r & (LSTM_H - 1);
    const float hv = h0[idx];
    const float cv = c0[idx];
    ph[(dir << 15) + (j >> 2) * 256 + ((j >> 1) & 1) * 128 + 2 * b + (j & 1)] = hv;
    ct[(dir << 15) + j * 64 + b] = cv;
}

// ---------------------------------------------------------------------------
// Fused per-timestep kernel.  grid = (16, 2): blockIdx.y = direction.
// 128 wave-slots/dir; wave: 16 batch rows x 16 hidden cols x 4 gates.
// K loop: [x_t (in_dim) | h_{t-1} (H)] against packed wcat, then in-register
// sigmoid/tanh cell update, c in place, h to the other packed buffer.
// ---------------------------------------------------------------------------
__global__ __launch_bounds__(256) void lstm_step_fused(
    const float* __restrict__ xsrc,      // raw input (layer0) or packed py0 (layer1)
    int x_packed, int x_bstride, int x_tstride,
    int in_dim, int ktot,
    int tf, int tb,
    const float* __restrict__ wpk_f, const float* __restrict__ wpk_b,
    const float* __restrict__ bias_f, const float* __restrict__ bias_b,
    const float* __restrict__ ph_in,     // packed h_{t-1} [2][H/4][2][B][2]
    float* __restrict__ ph_out,          // packed h_t
    float* __restrict__ ct,              // c state [2][H][B] (single-owner lanes)
    float* __restrict__ ypack, int have_ypack,   // layer0: packed py0
    float* __restrict__ ylast, int we_f, int we_b) // layer1: [2H][B] last capture
{
    const int dir = blockIdx.y;
    const int t = dir ? tb : tf;
    const float* wpk  = dir ? wpk_b  : wpk_f;
    const float* bias = dir ? bias_b : bias_f;

    const int lane  = threadIdx.x & 31;
    const int wslot = blockIdx.x * 8 + (threadIdx.x >> 5); // 0..127
    const int mtile = wslot >> 5;                          // 0..3
    const int jt    = wslot & 31;                          // 0..31 (H/16)
    const int lm    = lane & 15;
    const int h2    = lane >> 4;                           // K-pair half
    const int kh    = h2 << 1;
    const int row   = mtile * 16 + lm;                     // batch row (A frag)
    const int jcol  = jt * 16 + lm;                        // hidden column
    const int mbase = mtile * 16 + (h2 << 3);              // acc rows mbase..+7

    // Bias-seeded accumulators; acc[g] is gate g at column g*512+jcol.
    v8f acc[4];
#pragma unroll
    for (int g = 0; g < 4; ++g) {
        const float bv = bias[g * LSTM_H + jcol];
        v8f a0 = {bv, bv, bv, bv, bv, bv, bv, bv};
        acc[g] = a0;
    }

    // Packed-B base pointers (advance by k*2048 floats; coalesced v2f loads).
    const float* wb[4];
#pragma unroll
    for (int g = 0; g < 4; ++g)
        wb[g] = wpk + (size_t)h2 * 4096 + 2 * (g * LSTM_H + jt * 16 + lm);

    // ---- Phase 1: x_t contribution (K = 0 .. in_dim) ----
    if (x_packed) {
        const float* xa = xsrc + (size_t)t * 65536 + h2 * 128 + 2 * row;
        for (int k = 0; k < in_dim; k += 4) {
            v2f a = ld2(xa + (size_t)k * 64);
#pragma unroll
            for (int g = 0; g < 4; ++g) {
                v2f bm = ld2(wb[g] + (size_t)k * 2048);
                acc[g] = __builtin_amdgcn_wmma_f32_16x16x4_f32(
                    false, a, false, bm, (short)0, acc[g], false, false);
            }
        }
    } else {
        const float* xa = xsrc + (size_t)row * x_bstride + (size_t)t * x_tstride + kh;
        for (int k = 0; k < in_dim; k += 4) {
            v2f a = ld2(xa + k);
#pragma unroll
            for (int g = 0; g < 4; ++g) {
                v2f bm = ld2(wb[g] + (size_t)k * 2048);
                acc[g] = __builtin_amdgcn_wmma_f32_16x16x4_f32(
                    false, a, false, bm, (short)0, acc[g], false, false);
            }
        }
    }

    // ---- Phase 2: h_{t-1} contribution (K = in_dim .. ktot) ----
    const float* ha = ph_in + (size_t)(dir << 15) + h2 * 128 + 2 * row;
    for (int k = in_dim; k < ktot; k += 4) {
        v2f a = ld2(ha + (size_t)(k - in_dim) * 64);
#pragma unroll
        for (int g = 0; g < 4; ++g) {
            v2f bm = ld2(wb[g] + (size_t)k * 2048);
            acc[g] = __builtin_amdgcn_wmma_f32_16x16x4_f32(
                false, a, false, bm, (short)0, acc[g], false, false);
        }
    }

    // ---- Fused LSTM cell update (all four gates resident in registers) ----
    // Lane owns (rows mbase..mbase+7, column jcol) exclusively -> c in place.
    const size_t cbase = (size_t)(dir << 15) + (size_t)jcol * 64 + mbase;
    v8f cp = *reinterpret_cast<const v8f*>(ct + cbase);
    v8f cn, hv;
#pragma unroll
    for (int r = 0; r < 8; ++r) {
        const float si = 1.0f / (1.0f + __expf(-acc[0][r]));
        const float sf = 1.0f / (1.0f + __expf(-acc[1][r]));
        const float gg = tanhf(acc[2][r]);
        const float so = 1.0f / (1.0f + __expf(-acc[3][r]));
        const float c2 = sf * cp[r] + si * gg;
        cn[r] = c2;
        hv[r] = so * tanhf(c2);
    }
    *reinterpret_cast<v8f*>(ct + cbase) = cn;

    // h_t into the *other* packed buffer (race-free vs this step's readers).
    const int pb = (dir << 15) + (jcol >> 2) * 256 + ((jcol >> 1) & 1) * 128 + (jcol & 1);
#pragma unroll
    for (int r = 0; r < 8; ++r)
        ph_out[pb + 2 * (mbase + r)] = hv[r];

    if (have_ypack) {
        // Layer 0: emit time-aligned concat column (dir*H + jcol) in packed-A form.
        const int kcol = dir * LSTM_H + jcol;
        const size_t yb = (size_t)t * 65536 + (size_t)(kcol >> 2) * 256 +
                          ((kcol >> 1) & 1) * 128 + (kcol & 1);
#pragma unroll
        for (int r = 0; r < 8; ++r)
            ypack[yb + 2 * (mbase + r)] = hv[r];
    } else {
        const int we = dir ? we_b : we_f;
        if (we) {
#pragma unroll
            for (int r = 0; r < 8; ++r)
                ylast[(size_t)(dir * LSTM_H + jcol) * 64 + mbase + r] = hv[r];
        }
    }
}

// ---------------------------------------------------------------------------
// Final FC: out[b,o] = sum_k last[k][b] * fc_w[o][k] + fc_b[o]
// ---------------------------------------------------------------------------
__global__ __launch_bounds__(512) void lstm_fc(
    const float* __restrict__ last_t,   // [2H][B]
    const float* __restrict__ w,        // [O][2H]
    const float* __restrict__ bias,     // [O]
    float* __restrict__ out)            // [1][B][O]
{
    const int tid = threadIdx.x;        // 0..511 == b*8+o
    const int b = tid >> 3, o = tid & 7;
    const float* wr = w + (size_t)o * (2 * LSTM_H);
    float s = bias[o];
    for (int k = 0; k < 2 * LSTM_H; ++k) s += last_t[(size_t)k * 64 + b] * wr[k];
    out[tid] = s;
}

// ---------------------------------------------------------------------------
// Host orchestration (all on `stream`, graph-capturable, deterministic):
//   4x weight pack, then per layer: init h/c, 512x fused step, then FC.
// ---------------------------------------------------------------------------
extern "C" void kernel_launch(void* const* d_in, const int* in_sizes, int n_in,
                              void* d_out, int out_size, void* d_ws, size_t ws_size,
                              hipStream_t stream) {
    (void)in_sizes; (void)n_in; (void)out_size; (void)ws_size;

    const float* input = (const float*)d_in[0];  // [64,512,256]
    const float* h0    = (const float*)d_in[1];  // [4,64,512]
    const float* c0    = (const float*)d_in[2];  // [4,64,512]
    const float* fc_w  = (const float*)d_in[15]; // [8,1024]
    const float* fc_b  = (const float*)d_in[16]; // [8]

    // Workspace layout (floats):
    const size_t PY0_ELEMS = (size_t)LSTM_S * 65536;  // packed layer-0 output: 33,554,432
    const size_t WPK_L0    = (size_t)(LSTM_I + LSTM_H) * LSTM_G;      // 1,572,864
    const size_t WPK_L1    = (size_t)(2 * LSTM_H + LSTM_H) * LSTM_G;  // 3,145,728
    const size_t PH_ELEMS  = (size_t)2 * LSTM_B * LSTM_H;             // 65,536

    float* py0    = (float*)d_ws;            // packed [S][K/4][2][B][2]
    float* wpk_0f = py0 + PY0_ELEMS;
    float* wpk_0b = wpk_0f + WPK_L0;
    float* wpk_1f = wpk_0b + WPK_L0;
    float* wpk_1b = wpk_1f + WPK_L1;
    float* phA    = wpk_1b + WPK_L1;         // packed h, double buffered
    float* phB    = phA + PH_ELEMS;
    float* ct     = phB + PH_ELEMS;          // c [2][H][B]
    float* last_t = ct + PH_ELEMS;           // [2H][B]

    // ---- One-time per call: pack [W_ih ; W_hh] for all 4 (layer,dir) ----
    {
        const int kt0 = LSTM_I + LSTM_H;        // 768
        const int kt1 = 2 * LSTM_H + LSTM_H;    // 1536
        lstm_pack_w<<<(LSTM_G * kt0) / 256, 256, 0, stream>>>(
            (const float*)d_in[3], (const float*)d_in[4], LSTM_I, kt0, wpk_0f);
        lstm_pack_w<<<(LSTM_G * kt0) / 256, 256, 0, stream>>>(
            (const float*)d_in[6], (const float*)d_in[7], LSTM_I, kt0, wpk_0b);
        lstm_pack_w<<<(LSTM_G * kt1) / 256, 256, 0, stream>>>(
            (const float*)d_in[9], (const float*)d_in[10], 2 * LSTM_H, kt1, wpk_1f);
        lstm_pack_w<<<(LSTM_G * kt1) / 256, 256, 0, stream>>>(
            (const float*)d_in[12], (const float*)d_in[13], 2 * LSTM_H, kt1, wpk_1b);
    }

    for (int l = 0; l < 2; ++l) {
        lstm_init_hc<<<256, 256, 0, stream>>>(
            h0 + (size_t)l * PH_ELEMS, c0 + (size_t)l * PH_ELEMS, phA, ct);

        const int base = 3 + l * 6;
        const float* b_f = (const float*)d_in[base + 2];
        const float* b_b = (const float*)d_in[base + 5];
        const float* wpf = l ? wpk_1f : wpk_0f;
        const float* wpb = l ? wpk_1b : wpk_0b;

        const float* xsrc = l ? py0 : input;
        const int x_packed  = l ? 1 : 0;
        const int x_bstride = LSTM_S * LSTM_I;  // raw input: per batch row
        const int x_tstride = LSTM_I;           // raw input: per timestep
        const int in_dim    = l ? (2 * LSTM_H) : LSTM_I;
        const int ktot      = in_dim + LSTM_H;

        for (int t = 0; t < LSTM_S; ++t) {
            const int tf = t, tb = LSTM_S - 1 - t;
            const float* pin = (t & 1) ? phB : phA;
            float*       pout = (t & 1) ? phA : phB;

            if (l == 0) {
                lstm_step_fused<<<dim3(16, 2), 256, 0, stream>>>(
                    xsrc, x_packed, x_bstride, x_tstride, in_dim, ktot,
                    tf, tb, wpf, wpb, b_f, b_b, pin, pout, ct,
                    py0, 1, (float*)nullptr, 0, 0);
            } else {
                lstm_step_fused<<<dim3(16, 2), 256, 0, stream>>>(
                    xsrc, x_packed, x_bstride, x_tstride, in_dim, ktot,
                    tf, tb, wpf, wpb, b_f, b_b, pin, pout, ct,
                    (float*)nullptr, 0,
                    last_t, (t == LSTM_S - 1) ? 1 : 0, (t == 0) ? 1 : 0);
            }
        }
    }

    lstm_fc<<<1, 512, 0, stream>>>(last_t, fc_w, fc_b, (float*)d_out);
}